// GINEncoder_6786048328257
// MI455X (gfx1250) — compile-verified
//
#include <hip/hip_runtime.h>
#include <hip/hip_bf16.h>

// ---------------------------------------------------------------------------
// GIN encoder for MI455X (gfx1250, wave32, WMMA).
//
// Pipeline (all on `stream`):
//   0) zero pool+cnt
//   1) agg1 = (1+eps1) * x                          [12.8M f32, streaming]
//   2) agg1[dst] += x[src]  (1.6M edges x 128)       [global_atomic_add_f32]
//   3) h1 = mlp1(agg1); agg2 = (1+eps2)*h1           [WMMA f32 16x16x4, fused BN+ReLU]
//   4) agg2[dst] += h1[src] (1.6M edges x 64)        [global_atomic_add_f32]
//   5) h2 = mlp2(agg2); pool[g] += h2; cnt[g] += 1   [WMMA + run-compressed pool atomics]
//   6) out = pool / max(cnt,1)
//
// Weights live in LDS in a pair-interleaved layout: element (k,n) at
// (k>>1)*2N + 2n + (k&1), so a WMMA B-fragment {W[kk][n], W[kk+1][n]} is one
// aligned ds_load_b64 straight into a contiguous even VGPR pair (no repack movs).
// ---------------------------------------------------------------------------

#define N_NODES  100000
#define N_EDGES  1600000
#define N_GRAPHS 512
#define IN_DIM   128
#define HID      64
#define EMB      128
#define N_TILES  (N_NODES / 16)   // 6250, exact

typedef __attribute__((ext_vector_type(2))) float v2f;
typedef __attribute__((ext_vector_type(8))) float v8f;

__device__ __forceinline__ v8f wmma4(v2f a, v2f b, v8f c) {
  // D = A(16x4 f32) x B(4x16 f32) + C(16x16 f32), full f32 precision
  return __builtin_amdgcn_wmma_f32_16x16x4_f32(
      /*neg_a=*/false, a, /*neg_b=*/false, b,
      /*c_mod=*/(short)0, c, /*reuse_a=*/false, /*reuse_b=*/false);
}

__device__ __forceinline__ void atomic_add_f32(float* p, float v) {
  // relaxed agent-scope FP add -> hardware global_atomic_add_f32 (no CAS loop)
  __hip_atomic_fetch_add(p, v, __ATOMIC_RELAXED, __HIP_MEMORY_SCOPE_AGENT);
}

// ---------------------------------------------------------------------------
__global__ void zero_kernel(float* __restrict__ p, int n) {
  int i = blockIdx.x * blockDim.x + threadIdx.x;
  if (i < n) p[i] = 0.0f;
}

__global__ void scale_kernel(const float4* __restrict__ x, float4* __restrict__ out,
                             const float* __restrict__ epsp, int n4) {
  int i = blockIdx.x * blockDim.x + threadIdx.x;
  if (i >= n4) return;
  float s = 1.0f + epsp[0];
  float4 v = x[i];
  v.x *= s; v.y *= s; v.z *= s; v.w *= s;
  out[i] = v;
}

// One wave (32 lanes) per edge; lane covers 4 consecutive feats (128 total).
__global__ void scatter128_kernel(const float* __restrict__ h,
                                  const int* __restrict__ ei,
                                  float* __restrict__ agg) {
  unsigned tid = blockIdx.x * blockDim.x + threadIdx.x;
  unsigned e = tid >> 5, lane = tid & 31;
  if (e >= N_EDGES) return;
  int s = ei[e];
  int d = ei[N_EDGES + e];
  float4 v = ((const float4*)(h + (size_t)s * IN_DIM))[lane];
  float* p = agg + (size_t)d * IN_DIM + lane * 4;
  atomic_add_f32(p + 0, v.x); atomic_add_f32(p + 1, v.y);
  atomic_add_f32(p + 2, v.z); atomic_add_f32(p + 3, v.w);
}

// One wave per edge; lane covers 2 consecutive feats (64 total).
__global__ void scatter64_kernel(const float* __restrict__ h,
                                 const int* __restrict__ ei,
                                 float* __restrict__ agg) {
  unsigned tid = blockIdx.x * blockDim.x + threadIdx.x;
  unsigned e = tid >> 5, lane = tid & 31;
  if (e >= N_EDGES) return;
  int s = ei[e];
  int d = ei[N_EDGES + e];
  float2 v = ((const float2*)(h + (size_t)s * HID))[lane];
  float* p = agg + (size_t)d * HID + lane * 2;
  atomic_add_f32(p + 0, v.x); atomic_add_f32(p + 1, v.y);
}

// pair-interleaved LDS weight staging: (k,n) -> (k>>1)*2N + 2n + (k&1)
template <int K, int N>
__device__ __forceinline__ void stage_weights(float* sw, const float* __restrict__ W,
                                              int t, int nthreads) {
  for (int i = t; i < K * N; i += nthreads) {
    int k = i / N, n = i - k * N;                 // consecutive i -> coalesced reads
    sw[(k >> 1) * (2 * N) + 2 * n + (k & 1)] = W[i];
  }
}

// ---------------------------------------------------------------------------
// MLP1: [16x128] @ [128x64] -> BN+ReLU -> @ [64x64] -> BN+ReLU.
// 8 waves/block, one 16-row node tile per wave, WMMA f32 16x16x4.
__launch_bounds__(256)
__global__ void gin_mlp1_kernel(const float* __restrict__ agg,
    const float* __restrict__ W1, const float* __restrict__ W2,
    const float* __restrict__ b1, const float* __restrict__ b2,
    const float* __restrict__ bn1_beta, const float* __restrict__ bn1_gamma,
    const float* __restrict__ bn1_mean, const float* __restrict__ bn1_var,
    const float* __restrict__ bn2_beta, const float* __restrict__ bn2_gamma,
    const float* __restrict__ bn2_mean, const float* __restrict__ bn2_var,
    const float* __restrict__ eps2p,
    float* __restrict__ h1, float* __restrict__ agg2) {
  __shared__ float sW1[IN_DIM * HID];      // 32 KB, pair-interleaved
  __shared__ float sW2[HID * HID];         // 16 KB, pair-interleaved
  __shared__ float s1[HID], u1[HID], s2[HID], u2[HID];
  __shared__ float sMid[8 * 16 * HID];     // 32 KB (per-wave 16x64 slices)

  int t = threadIdx.x;
  stage_weights<IN_DIM, HID>(sW1, W1, t, 256);
  stage_weights<HID, HID>(sW2, W2, t, 256);
  if (t < HID) {                  // fold bias + BN into y = c*s + u
    float s = bn1_gamma[t] * rsqrtf(bn1_var[t] + 1e-5f);
    s1[t] = s; u1[t] = fmaf(b1[t] - bn1_mean[t], s, bn1_beta[t]);
  } else if (t < 2 * HID) {
    int n = t - HID;
    float s = bn2_gamma[n] * rsqrtf(bn2_var[n] + 1e-5f);
    s2[n] = s; u2[n] = fmaf(b2[n] - bn2_mean[n], s, bn2_beta[n]);
  }
  __syncthreads();

  int wave = t >> 5, lane = t & 31;
  int tile = blockIdx.x * 8 + wave;
  if (tile >= N_TILES) return;            // whole wave exits -> EXEC all-1 for WMMA
  int M = lane & 15, hi = lane >> 4, n16 = lane & 15;

  // ---- GEMM1: A[16x128] x W1[128x64] ----
  const float* A = agg + (size_t)tile * 16 * IN_DIM;
  v8f acc[4] = {};
  #pragma unroll 4
  for (int k0 = 0; k0 < IN_DIM; k0 += 4) {
    int kk = k0 + 2 * hi;                 // A layout: lane=M, comps K={kk,kk+1}
    v2f a = *(const v2f*)(A + M * IN_DIM + kk);
    const float* brow = sW1 + ((k0 >> 1) + hi) * (2 * HID);  // pair (kk,kk+1)
    #pragma unroll
    for (int tt = 0; tt < 4; ++tt) {      // B layout: lane=N; one ds_load_b64
      v2f b = *(const v2f*)(brow + 2 * (tt * 16 + n16));
      acc[tt] = wmma4(a, b, acc[tt]);
    }
  }
  // epilogue1 (BN+ReLU) -> LDS for GEMM2 A-operand
  float* mid = sMid + wave * 16 * HID;
  #pragma unroll
  for (int tt = 0; tt < 4; ++tt) {
    int n = tt * 16 + n16;
    float s = s1[n], u = u1[n];
    #pragma unroll
    for (int j = 0; j < 8; ++j) {         // C layout: comp j <-> row j + 8*hi
      float c = fmaxf(fmaf(acc[tt][j], s, u), 0.0f);
      mid[(j + 8 * hi) * HID + n] = c;
    }
  }
  asm volatile("s_wait_dscnt 0" ::: "memory");   // wave-internal LDS handoff
  __builtin_amdgcn_wave_barrier();

  // ---- GEMM2: mid[16x64] x W2[64x64] ----
  v8f d[4] = {};
  #pragma unroll 4
  for (int k0 = 0; k0 < HID; k0 += 4) {
    int kk = k0 + 2 * hi;
    v2f a = *(const v2f*)(mid + M * HID + kk);   // contiguous in k -> ds_load_b64
    const float* brow = sW2 + ((k0 >> 1) + hi) * (2 * HID);
    #pragma unroll
    for (int tt = 0; tt < 4; ++tt) {
      v2f b = *(const v2f*)(brow + 2 * (tt * 16 + n16));
      d[tt] = wmma4(a, b, d[tt]);
    }
  }
  float e2 = 1.0f + eps2p[0];
  #pragma unroll
  for (int tt = 0; tt < 4; ++tt) {
    int n = tt * 16 + n16;
    float s = s2[n], u = u2[n];
    #pragma unroll
    for (int j = 0; j < 8; ++j) {
      int row = tile * 16 + j + 8 * hi;
      float c = fmaxf(fmaf(d[tt][j], s, u), 0.0f);
      h1[(size_t)row * HID + n]   = c;        // scatter source for layer 2
      agg2[(size_t)row * HID + n] = e2 * c;   // pre-initialized self term
    }
  }
}

// ---------------------------------------------------------------------------
// MLP2: [16x64] @ [64x64] -> BN+ReLU -> @ [64x128] -> BN+ReLU -> mean-pool atomics.
__launch_bounds__(256)
__global__ void gin_mlp2_kernel(const float* __restrict__ agg2,
    const float* __restrict__ W1, const float* __restrict__ W2,
    const float* __restrict__ b1, const float* __restrict__ b2,
    const float* __restrict__ bn1_beta, const float* __restrict__ bn1_gamma,
    const float* __restrict__ bn1_mean, const float* __restrict__ bn1_var,
    const float* __restrict__ bn2_beta, const float* __restrict__ bn2_gamma,
    const float* __restrict__ bn2_mean, const float* __restrict__ bn2_var,
    const int* __restrict__ batch_vec,
    float* __restrict__ pool, float* __restrict__ cnt) {
  __shared__ float sW1[HID * HID];         // 16 KB, pair-interleaved
  __shared__ float sW2[HID * EMB];         // 32 KB, pair-interleaved
  __shared__ float s1[HID], u1[HID], s2[EMB], u2[EMB];
  __shared__ float sMid[8 * 16 * HID];     // 32 KB

  int t = threadIdx.x;
  stage_weights<HID, HID>(sW1, W1, t, 256);
  stage_weights<HID, EMB>(sW2, W2, t, 256);
  if (t < HID) {
    float s = bn1_gamma[t] * rsqrtf(bn1_var[t] + 1e-5f);
    s1[t] = s; u1[t] = fmaf(b1[t] - bn1_mean[t], s, bn1_beta[t]);
  } else if (t < HID + EMB) {
    int n = t - HID;
    float s = bn2_gamma[n] * rsqrtf(bn2_var[n] + 1e-5f);
    s2[n] = s; u2[n] = fmaf(b2[n] - bn2_mean[n], s, bn2_beta[n]);
  }
  __syncthreads();

  int wave = t >> 5, lane = t & 31;
  int tile = blockIdx.x * 8 + wave;
  if (tile >= N_TILES) return;
  int M = lane & 15, hi = lane >> 4, n16 = lane & 15;

  // ---- GEMM1: A[16x64] x W1[64x64] ----
  const float* A = agg2 + (size_t)tile * 16 * HID;
  v8f acc[4] = {};
  #pragma unroll 4
  for (int k0 = 0; k0 < HID; k0 += 4) {
    int kk = k0 + 2 * hi;
    v2f a = *(const v2f*)(A + M * HID + kk);
    const float* brow = sW1 + ((k0 >> 1) + hi) * (2 * HID);
    #pragma unroll
    for (int tt = 0; tt < 4; ++tt) {
      v2f b = *(const v2f*)(brow + 2 * (tt * 16 + n16));
      acc[tt] = wmma4(a, b, acc[tt]);
    }
  }
  float* mid = sMid + wave * 16 * HID;
  #pragma unroll
  for (int tt = 0; tt < 4; ++tt) {
    int n = tt * 16 + n16;
    float s = s1[n], u = u1[n];
    #pragma unroll
    for (int j = 0; j < 8; ++j) {
      float c = fmaxf(fmaf(acc[tt][j], s, u), 0.0f);
      mid[(j + 8 * hi) * HID + n] = c;
    }
  }
  asm volatile("s_wait_dscnt 0" ::: "memory");
  __builtin_amdgcn_wave_barrier();

  // ---- GEMM2: mid[16x64] x W2[64x128] ----
  v8f d[8] = {};
  #pragma unroll 4
  for (int k0 = 0; k0 < HID; k0 += 4) {
    int kk = k0 + 2 * hi;
    v2f a = *(const v2f*)(mid + M * HID + kk);
    const float* brow = sW2 + ((k0 >> 1) + hi) * (2 * EMB);
    #pragma unroll
    for (int tt = 0; tt < 8; ++tt) {
      v2f b = *(const v2f*)(brow + 2 * (tt * 16 + n16));
      d[tt] = wmma4(a, b, d[tt]);
    }
  }

  // graph ids of the 8 rows this lane-half owns (batch_vec is sorted)
  int gg[8];
  #pragma unroll
  for (int j = 0; j < 8; ++j) gg[j] = batch_vec[tile * 16 + j + 8 * hi];

  // epilogue2 + run-length-compressed pooled atomics
  #pragma unroll
  for (int tt = 0; tt < 8; ++tt) {
    int n = tt * 16 + n16;
    float s = s2[n], u = u2[n];
    float accv = 0.0f; int curg = gg[0];
    #pragma unroll
    for (int j = 0; j < 8; ++j) {
      float c = fmaxf(fmaf(d[tt][j], s, u), 0.0f);
      if (gg[j] != curg) {
        atomic_add_f32(&pool[(size_t)curg * EMB + n], accv);
        accv = 0.0f; curg = gg[j];
      }
      accv += c;
    }
    atomic_add_f32(&pool[(size_t)curg * EMB + n], accv);
  }
  if ((lane & 15) == 0) {                  // lane 0 -> rows 0..7, lane 16 -> rows 8..15
    float cacc = 0.0f; int curg = gg[0];
    #pragma unroll
    for (int j = 0; j < 8; ++j) {
      if (gg[j] != curg) { atomic_add_f32(&cnt[curg], cacc); cacc = 0.0f; curg = gg[j]; }
      cacc += 1.0f;
    }
    atomic_add_f32(&cnt[curg], cacc);
  }
}

__global__ void finalize_kernel(const float* __restrict__ pool,
                                const float* __restrict__ cnt,
                                float* __restrict__ out) {
  int i = blockIdx.x * blockDim.x + threadIdx.x;
  if (i >= N_GRAPHS * EMB) return;
  int g = i / EMB;
  out[i] = pool[i] / fmaxf(cnt[g], 1.0f);
}

// ---------------------------------------------------------------------------
extern "C" void kernel_launch(void* const* d_in, const int* in_sizes, int n_in,
                              void* d_out, int out_size, void* d_ws, size_t ws_size,
                              hipStream_t stream) {
  // setup_inputs() order, params flattened as a jax pytree (dict keys sorted):
  // 0:x 1:edge_index 2:edge_weight(unused) 3:batch_vec 4:eps1 5:eps2
  // mlp1: 6:W1 7:W2 8:b1 9:b2 10..13:bn1{beta,gamma,mean,var} 14..17:bn2{...}
  // mlp2: 18:W1 19:W2 20:b1 21:b2 22..25:bn1{...} 26..29:bn2{...}
  const float* x    = (const float*)d_in[0];
  const int* ei     = (const int*)d_in[1];
  const int* batch  = (const int*)d_in[3];
  const float* eps1 = (const float*)d_in[4];
  const float* eps2 = (const float*)d_in[5];
  const float* m1W1 = (const float*)d_in[6];
  const float* m1W2 = (const float*)d_in[7];
  const float* m1b1 = (const float*)d_in[8];
  const float* m1b2 = (const float*)d_in[9];
  const float* m1bn1_beta = (const float*)d_in[10], *m1bn1_gamma = (const float*)d_in[11];
  const float* m1bn1_mean = (const float*)d_in[12], *m1bn1_var   = (const float*)d_in[13];
  const float* m1bn2_beta = (const float*)d_in[14], *m1bn2_gamma = (const float*)d_in[15];
  const float* m1bn2_mean = (const float*)d_in[16], *m1bn2_var   = (const float*)d_in[17];
  const float* m2W1 = (const float*)d_in[18];
  const float* m2W2 = (const float*)d_in[19];
  const float* m2b1 = (const float*)d_in[20];
  const float* m2b2 = (const float*)d_in[21];
  const float* m2bn1_beta = (const float*)d_in[22], *m2bn1_gamma = (const float*)d_in[23];
  const float* m2bn1_mean = (const float*)d_in[24], *m2bn1_var   = (const float*)d_in[25];
  const float* m2bn2_beta = (const float*)d_in[26], *m2bn2_gamma = (const float*)d_in[27];
  const float* m2bn2_mean = (const float*)d_in[28], *m2bn2_var   = (const float*)d_in[29];

  // workspace layout (f32): agg1 | h1 | agg2 | pool | cnt
  float* agg1 = (float*)d_ws;
  float* h1v  = agg1 + (size_t)N_NODES * IN_DIM;
  float* agg2 = h1v  + (size_t)N_NODES * HID;
  float* pool = agg2 + (size_t)N_NODES * HID;
  float* cntv = pool + (size_t)N_GRAPHS * EMB;

  // 0) zero pool + cnt (contiguous)
  {
    int n = N_GRAPHS * EMB + N_GRAPHS;
    zero_kernel<<<(n + 255) / 256, 256, 0, stream>>>(pool, n);
  }
  // 1) agg1 = (1+eps1)*x
  {
    int n4 = N_NODES * IN_DIM / 4;
    scale_kernel<<<(n4 + 255) / 256, 256, 0, stream>>>(
        (const float4*)x, (float4*)agg1, eps1, n4);
  }
  // 2) edge scatter, 128 feats (wave per edge)
  scatter128_kernel<<<(N_EDGES * 32) / 256, 256, 0, stream>>>(x, ei, agg1);
  // 3) mlp1 (WMMA), also writes agg2 = (1+eps2)*h1
  gin_mlp1_kernel<<<(N_TILES + 7) / 8, 256, 0, stream>>>(
      agg1, m1W1, m1W2, m1b1, m1b2,
      m1bn1_beta, m1bn1_gamma, m1bn1_mean, m1bn1_var,
      m1bn2_beta, m1bn2_gamma, m1bn2_mean, m1bn2_var,
      eps2, h1v, agg2);
  // 4) edge scatter, 64 feats
  scatter64_kernel<<<(N_EDGES * 32) / 256, 256, 0, stream>>>(h1v, ei, agg2);
  // 5) mlp2 (WMMA) + pooled atomics
  gin_mlp2_kernel<<<(N_TILES + 7) / 8, 256, 0, stream>>>(
      agg2, m2W1, m2W2, m2b1, m2b2,
      m2bn1_beta, m2bn1_gamma, m2bn1_mean, m2bn1_var,
      m2bn2_beta, m2bn2_gamma, m2bn2_mean, m2bn2_var,
      batch, pool, cntv);
  // 6) out = pool / max(cnt,1)
  finalize_kernel<<<(N_GRAPHS * EMB + 255) / 256, 256, 0, stream>>>(
      pool, cntv, (float*)d_out);
  (void)in_sizes; (void)n_in; (void)out_size; (void)ws_size;
}